// PostProcess_25177098289392
// MI455X (gfx1250) — compile-verified
//
#include <hip/hip_runtime.h>
#include <hip/hip_bf16.h>

// Problem constants (from reference)
#define TEXT_VOCAB 50265
#define VDIM       51267
#define BATCH      16
#define SEQL       256
#define NQ         100
#define NP         256

// Streaming layout for the argmax kernel
#define CHUNK      4096   // floats staged per workgroup per chunk
#define WAVE_ELEMS 512    // floats per wave per chunk (2 KB)
#define NFULL      12     // 12 * 4096 = 49152 full-chunk floats
#define TAIL_BASE  49152  // remaining 2115 handled directly

typedef float f32x4 __attribute__((ext_vector_type(4)));
typedef int   v4i   __attribute__((vector_size(16)));   // matches builtin param type

typedef __attribute__((address_space(1))) v4i* gv4i_ptr;
typedef __attribute__((address_space(3))) v4i* lv4i_ptr;

__device__ __forceinline__ void async_copy_b128(const float* gsrc, f32x4* lds_dst) {
    // CDNA5: GLOBAL_LOAD_ASYNC_TO_LDS_B128 — tracked by ASYNCcnt
    __builtin_amdgcn_global_load_async_to_lds_b128(
        (gv4i_ptr)(gsrc),
        (lv4i_ptr)(lds_dst),
        /*offset=*/0, /*cpol=*/0);
}

__global__ __launch_bounds__(256) void argmax_classify_kernel(
        const float* __restrict__ logits, int* __restrict__ codes) {
    __shared__ f32x4 lbuf[2][CHUNK / 4];   // 2 x 16 KB double buffer
    __shared__ float swval[8];
    __shared__ int   swidx[8];

    const int tid  = threadIdx.x;
    const int lane = tid & 31;
    const int wave = tid >> 5;
    const float* rp = logits + (size_t)blockIdx.x * (size_t)VDIM;

    float bval = -3.402823466e38f;
    int   bidx = 0x7FFFFFFF;

    // Prime the pipeline: issue chunk 0 into buffer 0 (4 async b128 per wave)
    {
        const float* g = rp + wave * WAVE_ELEMS + lane * 4;
        f32x4* l = &lbuf[0][wave * 128 + lane];
        #pragma unroll
        for (int i = 0; i < 4; ++i)
            async_copy_b128(g + i * 128, l + i * 32);
    }

    for (int c = 0; c < NFULL; ++c) {
        if (c + 1 < NFULL) {
            // Issue next chunk into the other buffer, then wait for current
            const float* g = rp + (c + 1) * CHUNK + wave * WAVE_ELEMS + lane * 4;
            f32x4* l = &lbuf[(c + 1) & 1][wave * 128 + lane];
            #pragma unroll
            for (int i = 0; i < 4; ++i)
                async_copy_b128(g + i * 128, l + i * 32);
            // 4 newer loads outstanding -> waiting <=4 guarantees chunk c landed
            asm volatile("s_wait_asynccnt 4" ::: "memory");
        } else {
            asm volatile("s_wait_asynccnt 0" ::: "memory");
        }

        const int gbase0 = c * CHUNK + wave * WAVE_ELEMS + lane * 4;
        #pragma unroll
        for (int i = 0; i < 4; ++i) {
            f32x4 v = lbuf[c & 1][wave * 128 + i * 32 + lane];  // ds_load_b128
            const int gb = gbase0 + i * 128;
            #pragma unroll
            for (int j = 0; j < 4; ++j) {
                float x = v[j];
                int   gi = gb + j;
                if (x > bval || (x == bval && gi < bidx)) { bval = x; bidx = gi; }
            }
        }
    }

    // Tail: 2115 leftover elements, nontemporal scalar loads
    for (int g = TAIL_BASE + tid; g < VDIM; g += 256) {
        float x = __builtin_nontemporal_load(rp + g);
        if (x > bval || (x == bval && g < bidx)) { bval = x; bidx = g; }
    }

    // Wave32 butterfly reduction (all lanes stay valid)
    #pragma unroll
    for (int off = 16; off > 0; off >>= 1) {
        float ov = __shfl_xor(bval, off, 32);
        int   oi = __shfl_xor(bidx, off, 32);
        if (ov > bval || (ov == bval && oi < bidx)) { bval = ov; bidx = oi; }
    }
    if (lane == 0) { swval[wave] = bval; swidx[wave] = bidx; }
    __syncthreads();

    if (tid == 0) {
        float fv = swval[0]; int fi = swidx[0];
        #pragma unroll
        for (int w = 1; w < 8; ++w) {
            float ov = swval[w]; int oi = swidx[w];
            if (ov > fv || (ov == fv && oi < fi)) { fv = ov; fi = oi; }
        }
        int code;
        if (fi < TEXT_VOCAB)            code = 0;  // text
        else if (fi == TEXT_VOCAB)      code = 1;  // open
        else if (fi == TEXT_VOCAB + 1)  code = 2;  // close
        else                            code = 3;  // bbox
        codes[blockIdx.x] = code;
    }
}

__global__ __launch_bounds__(128) void finalize_kernel(
        const int* __restrict__ codes, const float* __restrict__ boxes_in,
        const float* __restrict__ tsz, float* __restrict__ out) {
    __shared__ int cnt[NQ];
    __shared__ int lastf[NQ];
    const int b = blockIdx.x;
    const int t = threadIdx.x;

    if (t < NQ) { cnt[t] = 0; lastf[t] = 0; }
    __syncthreads();

    if (t == 0) {
        // Pass 1: total bbox tokens -> n_rows
        int total = 0;
        for (int l = 0; l < SEQL; ++l) total += (codes[b * SEQL + l] == 3);
        const int n_rows = (total + 3) >> 2;

        // Pass 2: sequential cumsums, record unique (row,col) scatter hits
        int in_bbox = 0, bcnt = 0, tpos = 0;
        for (int l = 0; l < SEQL; ++l) {
            const int c = codes[b * SEQL + l];
            in_bbox += (c == 1) - (c == 2);       // inclusive cumsum of delta
            const int bexcl = bcnt; bcnt += (c == 3);
            const int tp = tpos;    tpos += (c == 0);
            if (c == 0 && in_bbox != 0) {
                const int bi = bexcl >> 2;
                if (bi < n_rows) {
                    const int col = tp + 1;
                    if (col < NP) {
                        cnt[bi] += 1;
                        if (col == NP - 1) lastf[bi] = 1;
                    }
                }
            }
        }
    }
    __syncthreads();

    if (t < NQ) {
        const float E1 = 2.71828182845904523536f;
        const int k = cnt[t];
        const float denom = 256.0f + (float)k * (E1 - 1.0f);
        const float pl = (lastf[t] ? E1 : 1.0f) / denom;
        const int oidx = b * NQ + t;

        out[oidx] = 1.0f - pl;                    // scores (16,100)
        out[BATCH * NQ + oidx] = 1.0f;            // labels (16,100) == 1

        const float cx = boxes_in[oidx * 4 + 0];
        const float cy = boxes_in[oidx * 4 + 1];
        const float w  = boxes_in[oidx * 4 + 2];
        const float h  = boxes_in[oidx * 4 + 3];
        const float ih = tsz[b * 2 + 0];
        const float iw = tsz[b * 2 + 1];
        float* ob = out + 2 * BATCH * NQ + oidx * 4;  // boxes (16,100,4)
        ob[0] = (cx - 0.5f * w) * iw;
        ob[1] = (cy - 0.5f * h) * ih;
        ob[2] = (cx + 0.5f * w) * iw;
        ob[3] = (cy + 0.5f * h) * ih;
    }
}

extern "C" void kernel_launch(void* const* d_in, const int* in_sizes, int n_in,
                              void* d_out, int out_size, void* d_ws, size_t ws_size,
                              hipStream_t stream) {
    const float* logits = (const float*)d_in[0];   // (16,256,51267) f32
    const float* pboxes = (const float*)d_in[1];   // (16,100,4)     f32
    const float* tsz    = (const float*)d_in[2];   // (16,2)         f32
    float* out = (float*)d_out;                    // 1600 + 1600 + 6400 floats
    int* codes = (int*)d_ws;                       // 4096 ints of scratch

    argmax_classify_kernel<<<BATCH * SEQL, 256, 0, stream>>>(logits, codes);
    finalize_kernel<<<BATCH, 128, 0, stream>>>(codes, pboxes, tsz, out);
}